// SPGG_Qlearning_51788715655333
// MI455X (gfx1250) — compile-verified
//
#include <hip/hip_runtime.h>

#define L      4096
#define LMASK  4095
#define ALPHA  0.8f
#define GAMMA  0.8f
#define EPS    0.02f

// tile geometry: 32x16 cells per block (1 cell/lane, 16 wave32s), halo of 2.
// Halo ratio 720/512 = 1.41x vs 1.69x for 32x8 -> less type-matrix HBM traffic.
#define TX 32
#define TY 16
#define NTHREADS (TX * TY)   // 512
#define HW (TX + 4)          // 36
#define HH (TY + 4)          // 20
#define HALO_N (HW * HH)     // 720

typedef float v4f __attribute__((ext_vector_type(4)));
typedef float v2f __attribute__((ext_vector_type(2)));

__global__ __launch_bounds__(NTHREADS) void spgg_qlearn_kernel(
    const int*   __restrict__ type_t,   // [L*L] int32 in {0,1}
    const float* __restrict__ Q,        // [N,2,2]
    const float* __restrict__ rnd,      // [N,2]
    const int*   __restrict__ rtype,    // [L*L]
    const float* __restrict__ mrand,    // [L*L]
    float* __restrict__ out_type,       // [N]
    float* __restrict__ out_Q,          // [N,2,2]
    float* __restrict__ out_profit)     // [N]
{
    __shared__ int tile[HALO_N];        // raw type values incl. torus halo

    const int tx  = threadIdx.x;
    const int ty  = threadIdx.y;
    const int tid = ty * TX + tx;
    const int bx  = blockIdx.x * TX;
    const int by  = blockIdx.y * TY;

    // LDS byte address of tile[]: generic LDS pointers carry the aperture in
    // the high 32 bits; low 32 bits are the workgroup-relative LDS offset.
    unsigned ldsBase = (unsigned)(uintptr_t)(void*)tile;

    // --- async halo fill: ASYNCcnt path, data never touches VGPRs ---
    for (int idx = tid; idx < HALO_N; idx += NTHREADS) {
        int ly = idx / HW;
        int lx = idx - ly * HW;
        int gy = (by + ly - 2) & LMASK;
        int gx = (bx + lx - 2) & LMASK;
        unsigned goff  = ((unsigned)((gy << 12) + gx)) * 4u;  // byte offset (GVS mode)
        unsigned laddr = ldsBase + (unsigned)idx * 4u;        // LDS byte addr
        asm volatile("global_load_async_to_lds_b32 %0, %1, %2"
                     :
                     : "v"(laddr), "v"(goff), "s"(type_t)
                     : "memory");
    }
    asm volatile("s_wait_asynccnt 0" ::: "memory");
    __syncthreads();

    // --- 13-point collapsed double stencil (integer taps, exact 0..25) ---
    const int cx = tx + 2, cy = ty + 2;
#define T(dy, dx) tile[(cy + (dy)) * HW + (cx + (dx))]
    const int C  = T(0, 0);
    const int S2 = 5 * C
                 + 2 * (T(0,-1) + T(0,1) + T(-1,0) + T(1,0)
                      + T(-1,-1) + T(-1,1) + T(1,-1) + T(1,1))
                 +      (T(0,-2) + T(0,2) + T(-2,0) + T(2,0));
#undef T
    // profit = (R/5)*S2 - 5*c  with R=4 -> 0.8*S2 - 5*c
    const float profit = 0.8f * (float)S2 - 5.0f * (float)C;

    const int y = by + ty;
    const int x = bx + tx;
    const int i = (y << 12) + x;

    // --- stream-once inputs: non-temporal, lane-consecutive b128 on Q ---
    v4f   q  = __builtin_nontemporal_load((const v4f*)Q + i);
    v2f   r  = __builtin_nontemporal_load((const v2f*)rnd + i);
    float mr = __builtin_nontemporal_load(mrand + i);
    int   rt = __builtin_nontemporal_load(rtype + i);

    // epsilon-greedy with random tie-break (argmax takes first on exact tie)
    const int   t  = C;                       // type in {0,1}
    const float q0 = t ? q.z : q.x;           // Q[i, t, 0]
    const float q1 = t ? q.w : q.y;           // Q[i, t, 1]
    const float s0 = (q0 >= q1 ? 1.0f : -1e9f) + r.x;
    const float s1 = (q1 >= q0 ? 1.0f : -1e9f) + r.y;
    const int greedy = (s1 > s0) ? 1 : 0;
    const int t1 = (mr >= EPS) ? greedy : rt;

    // TD update of Q[i, t, t1]
    const float n0 = t1 ? q.z : q.x;          // Q[i, t1, 0]
    const float n1 = t1 ? q.w : q.y;          // Q[i, t1, 1]
    const float max_next = fmaxf(n0, n1);
    const int   k    = 2 * t + t1;
    const float q_ab = (k == 0) ? q.x : (k == 1) ? q.y : (k == 2) ? q.z : q.w;
    const float nq   = q_ab + ALPHA * (profit + GAMMA * max_next - q_ab);
    if      (k == 0) q.x = nq;
    else if (k == 1) q.y = nq;
    else if (k == 2) q.z = nq;
    else             q.w = nq;

    // --- streaming outputs: NT stores keep the type matrix resident in L2 ---
    __builtin_nontemporal_store((float)t1, out_type + i);
    __builtin_nontemporal_store(q, (v4f*)out_Q + i);
    __builtin_nontemporal_store(profit, out_profit + i);
}

extern "C" void kernel_launch(void* const* d_in, const int* in_sizes, int n_in,
                              void* d_out, int out_size, void* d_ws, size_t ws_size,
                              hipStream_t stream) {
    const int*   type_t = (const int*)  d_in[0];
    const float* Q      = (const float*)d_in[1];
    const float* rnd    = (const float*)d_in[2];
    const int*   rtype  = (const int*)  d_in[3];
    const float* mrand  = (const float*)d_in[4];

    float* out = (float*)d_out;
    const size_t N = (size_t)L * L;
    // d_out layout (reference return order): [type_t1 | Q_new | profit]
    float* out_type   = out;
    float* out_Q      = out + N;
    float* out_profit = out + 5 * N;

    dim3 block(TX, TY);
    dim3 grid(L / TX, L / TY);
    spgg_qlearn_kernel<<<grid, block, 0, stream>>>(
        type_t, Q, rnd, rtype, mrand, out_type, out_Q, out_profit);
}